// LaplacianVectorLoss_50130858279309
// MI455X (gfx1250) — compile-verified
//
#include <hip/hip_runtime.h>

// ---------------------------------------------------------------------------
// Laplacian finite-difference edge scatter for MI455X (gfx1250).
// - Edge streams (src, dst, attr) are prefetched into LDS with the CDNA5
//   async-to-LDS DMA path (ASYNCcnt / s_wait_asynccnt), double buffered.
// - x[] gathers are L2-resident (8MB << 192MB L2).
// - Scatter uses native non-returning global_atomic_add_f32 (relaxed, agent
//   scope) so each edge costs two fire-and-forget L2 atomics.
// ---------------------------------------------------------------------------

constexpr int   BLOCK            = 256;   // 8 wave32s
constexpr int   CHUNK            = 256;   // edges per pipeline stage
constexpr int   CHUNKS_PER_BLOCK = 8;     // 2048 edges per block
constexpr float INV_DX2          = 10000.0f;  // 1 / (0.01^2)

__device__ __forceinline__ unsigned lds_addr_of(const void* p) {
  // Low 32 bits of a flat pointer into LDS == wave-relative LDS byte offset,
  // which is what the async-DMA VDST operand wants (an LDS byte offset).
  return (unsigned)(uintptr_t)p;
}

__device__ __forceinline__ void atomic_add_f32(float* p, float v) {
  // Relaxed, device(agent)-scope, result discarded -> non-returning
  // global_atomic_add_f32 (no CAS loop).
  (void)__hip_atomic_fetch_add(p, v, __ATOMIC_RELAXED, __HIP_MEMORY_SCOPE_AGENT);
}

__global__ __launch_bounds__(BLOCK) void laplacian_edge_kernel(
    const float2* __restrict__ x,
    const int*    __restrict__ edge_src,
    const int*    __restrict__ edge_dst,
    const float2* __restrict__ edge_attr,
    float* __restrict__ o_d2x,  float* __restrict__ o_d2xy,
    float* __restrict__ o_d2y,  float* __restrict__ o_d2yx,
    int total_chunks) {
  __shared__ int    s_src[2][CHUNK];
  __shared__ int    s_dst[2][CHUNK];
  __shared__ float2 s_att[2][CHUNK];

  const int tid    = threadIdx.x;
  const int chunk0 = blockIdx.x * CHUNKS_PER_BLOCK;
  int nch = total_chunks - chunk0;
  if (nch <= 0) return;
  if (nch > CHUNKS_PER_BLOCK) nch = CHUNKS_PER_BLOCK;

  // Async DMA prefetch of one chunk's edge data into LDS (3 async ops / wave).
  auto prefetch = [&](int buf, int chunk) {
    unsigned e    = (unsigned)chunk * CHUNK + (unsigned)tid;
    unsigned off4 = e * 4u;   // byte offset into int32 streams
    unsigned off8 = e * 8u;   // byte offset into float2 stream
    unsigned lsrc = lds_addr_of(&s_src[buf][tid]);
    unsigned ldst = lds_addr_of(&s_dst[buf][tid]);
    unsigned latt = lds_addr_of(&s_att[buf][tid]);
    asm volatile("global_load_async_to_lds_b32 %0, %1, %2"
                 :: "v"(lsrc), "v"(off4), "s"(edge_src) : "memory");
    asm volatile("global_load_async_to_lds_b32 %0, %1, %2"
                 :: "v"(ldst), "v"(off4), "s"(edge_dst) : "memory");
    asm volatile("global_load_async_to_lds_b64 %0, %1, %2"
                 :: "v"(latt), "v"(off8), "s"(edge_attr) : "memory");
  };

  prefetch(0, chunk0);

  for (int c = 0; c < nch; ++c) {
    if (c + 1 < nch) {
      prefetch((c + 1) & 1, chunk0 + c + 1);
      // Up to 6 async ops outstanding; wait until only the 3 newest remain ->
      // chunk c's data is resident in LDS (async loads complete in order).
      asm volatile("s_wait_asynccnt 0x3" ::: "memory");
    } else {
      asm volatile("s_wait_asynccnt 0x0" ::: "memory");
    }

    const int buf = c & 1;
    const int s = s_src[buf][tid];
    const int d = s_dst[buf][tid];
    const float2 a = s_att[buf][tid];

    const float2 xs = x[s];   // L2-resident gathers
    const float2 xd = x[d];
    const float dx0 = (xd.x - xs.x) * INV_DX2;
    const float dx1 = (xd.y - xs.y) * INV_DX2;

    // Masks are multiplicative in the reference; adding a masked 0.0f is a
    // no-op, so only touch the two live outputs per edge.
    if (a.x != 0.0f) {
      atomic_add_f32(&o_d2x[s],  dx0);   // xx operator on x-edges
      atomic_add_f32(&o_d2xy[s], dx1);   // yx operator on x-edges
    }
    if (a.y != 0.0f) {
      atomic_add_f32(&o_d2y[s],  dx1);   // yy operator on y-edges
      atomic_add_f32(&o_d2yx[s], dx0);   // xy operator on y-edges
    }
  }
}

// Tail kernel for E % CHUNK edges (plain loads; E=4M -> empty, kept for safety).
__global__ void laplacian_tail_kernel(
    const float2* __restrict__ x,
    const int*    __restrict__ edge_src,
    const int*    __restrict__ edge_dst,
    const float2* __restrict__ edge_attr,
    float* __restrict__ o_d2x,  float* __restrict__ o_d2xy,
    float* __restrict__ o_d2y,  float* __restrict__ o_d2yx,
    int start, int E) {
  int e = start + blockIdx.x * blockDim.x + threadIdx.x;
  if (e >= E) return;
  const int s = edge_src[e];
  const int d = edge_dst[e];
  const float2 a  = edge_attr[e];
  const float2 xs = x[s];
  const float2 xd = x[d];
  const float dx0 = (xd.x - xs.x) * INV_DX2;
  const float dx1 = (xd.y - xs.y) * INV_DX2;
  if (a.x != 0.0f) { atomic_add_f32(&o_d2x[s], dx0); atomic_add_f32(&o_d2xy[s], dx1); }
  if (a.y != 0.0f) { atomic_add_f32(&o_d2y[s], dx1); atomic_add_f32(&o_d2yx[s], dx0); }
}

__global__ void zero4_kernel(float4* __restrict__ out, int n4) {
  int i = blockIdx.x * blockDim.x + threadIdx.x;
  if (i < n4) out[i] = make_float4(0.f, 0.f, 0.f, 0.f);
}

__global__ void zero1_kernel(float* __restrict__ out, int n) {
  int i = blockIdx.x * blockDim.x + threadIdx.x;
  if (i < n) out[i] = 0.f;
}

extern "C" void kernel_launch(void* const* d_in, const int* in_sizes, int n_in,
                              void* d_out, int out_size, void* d_ws, size_t ws_size,
                              hipStream_t stream) {
  const float2* x  = (const float2*)d_in[0];
  const int*    ei = (const int*)d_in[1];      // [2, E] row-major: row0=src, row1=dst
  const float2* ea = (const float2*)d_in[2];   // [E, 2]

  const int N = in_sizes[0] / 2;
  const int E = in_sizes[1] / 2;

  float* out    = (float*)d_out;               // 4*N floats: d2x|d2xy|d2y|d2yx
  float* o_d2x  = out;
  float* o_d2xy = out + N;
  float* o_d2y  = out + 2 * N;
  float* o_d2yx = out + 3 * N;

  // Zero the outputs (harness poisons d_out).
  const int n4  = out_size / 4;
  const int rem = out_size - n4 * 4;
  if (n4 > 0)
    zero4_kernel<<<(n4 + BLOCK - 1) / BLOCK, BLOCK, 0, stream>>>((float4*)d_out, n4);
  if (rem > 0)
    zero1_kernel<<<1, BLOCK, 0, stream>>>(out + n4 * 4, rem);

  // Main async-pipelined edge pass over full chunks.
  const int total_chunks = E / CHUNK;
  if (total_chunks > 0) {
    const int blocks = (total_chunks + CHUNKS_PER_BLOCK - 1) / CHUNKS_PER_BLOCK;
    laplacian_edge_kernel<<<blocks, BLOCK, 0, stream>>>(
        x, ei, ei + E, ea, o_d2x, o_d2xy, o_d2y, o_d2yx, total_chunks);
  }

  // Tail (E % CHUNK edges).
  const int start = total_chunks * CHUNK;
  const int tail  = E - start;
  if (tail > 0) {
    laplacian_tail_kernel<<<(tail + BLOCK - 1) / BLOCK, BLOCK, 0, stream>>>(
        x, ei, ei + E, ea, o_d2x, o_d2xy, o_d2y, o_d2yx, start, E);
  }
}